// DotDetectionLoss_2310692405464
// MI455X (gfx1250) — compile-verified
//
#include <hip/hip_runtime.h>
#include <hip/hip_bf16.h>

// Problem constants (from reference): B=16, P=4096, T=1024
#define B_ 16
#define P_ 4096
#define T_ 1024
#define RADIUS_ 2.5f
#define R2_ 6.25f              // RADIUS^2: score>=0.5  <=>  d<=R  <=>  d2<=R2
#define LOG2E_ 1.44269504088896340736f

typedef __attribute__((ext_vector_type(2))) float v2f;
typedef __attribute__((ext_vector_type(8))) float v8f;

// ---------------------------------------------------------------------------
// Kernel 0: zero workspace (best keys + maxscore + correct flags)
// ---------------------------------------------------------------------------
__global__ void zero_ws_kernel(unsigned int* __restrict__ ws, int n) {
  int i = blockIdx.x * blockDim.x + threadIdx.x;
  if (i < n) ws[i] = 0u;
}

// ---------------------------------------------------------------------------
// Kernel 1: pairwise squared distances via V_WMMA_F32_16X16X4_F32.
//   d^2(p,t) = |p|^2 + |t|^2 - 2 p.t  ==  A(16x4) x B(4x16)
//   A row m = [px, py, |p|^2, 1],  B col n = [-2tx, -2ty, 1, |t|^2]
// One wave per (b, 16-pred tile); loops over all T in 16-target tiles.
// score = sigmoid(R - d) is strictly decreasing in d^2, so:
//   - per-pred max score  == per-pred MIN d^2 (sigmoid applied once at end)
//   - per-target argmax score == argmin d^2 (key from inverted d2 bits;
//     key magnitude is never consumed, only the packed pred index)
// Inner loop is fully branchless: invalid lanes contribute key==0 via cndmask.
// ---------------------------------------------------------------------------
__global__ void pair_kernel(const float* __restrict__ pred,  // [B][P][4] cls,x,y,conf
                            const float* __restrict__ gt,    // [B][T][3] cls,x,y
                            unsigned long long* __restrict__ best,     // [B][T]
                            float* __restrict__ maxscore)              // [B][P]
{
  const int lane = threadIdx.x & 31;
  const int idx  = lane & 15;   // column / row-within-16
  const int hf   = lane >> 4;   // half select
  const int wave = (blockIdx.x * blockDim.x + threadIdx.x) >> 5;
  const int b    = wave / (P_ / 16);
  const int p0   = (wave % (P_ / 16)) * 16;

  // --- A fragment (ISA 16x4 f32 layout: row M=idx for all lanes;
  //     v0 = K0 (lanes 0-15) / K2 (lanes 16-31); v1 = K1 / K3)
  const float* pr = pred + ((size_t)b * P_ + (size_t)(p0 + idx)) * 4;
  const float px = pr[1];
  const float py = pr[2];
  v2f afrag;
  if (hf == 0) { afrag.x = px;                afrag.y = py;   }
  else         { afrag.x = px * px + py * py; afrag.y = 1.0f; }

  // pred classes for the 8 D rows this lane sees (row m = r + 8*hf)
  float pcls[8];
  float rowmin[8];
#pragma unroll
  for (int r = 0; r < 8; ++r) {
    pcls[r]   = pred[((size_t)b * P_ + (size_t)(p0 + r + 8 * hf)) * 4 + 0];
    rowmin[r] = 3.4e38f;
  }
  // loop-invariant low half of the argmax key: ~p = 0xFFFFFFFF - (p0 + r + 8*hf)
  const unsigned keylo_base = 0xFFFFFFFFu - (unsigned)(p0 + 8 * hf);

  const float* gbase = gt + (size_t)b * T_ * 3;

  // Software-pipelined gt tile load (one tile ahead of the WMMA consuming it)
  float tcls = gbase[(size_t)idx * 3 + 0];
  float tx   = gbase[(size_t)idx * 3 + 1];
  float ty   = gbase[(size_t)idx * 3 + 2];

  for (int t0 = 0; t0 < T_; t0 += 16) {
    const float ctc = tcls, ctx = tx, cty = ty;
    if (t0 + 16 < T_) {
      const float* gr = gbase + (size_t)(t0 + 16 + idx) * 3;
      tcls = gr[0]; tx = gr[1]; ty = gr[2];
    }

    // --- B fragment (4x16: col N=idx for all lanes; v0 = K0/K2, v1 = K1/K3)
    v2f bfrag;
    if (hf == 0) { bfrag.x = -2.0f * ctx; bfrag.y = -2.0f * cty;           }
    else         { bfrag.x = 1.0f;        bfrag.y = ctx * ctx + cty * cty; }

    v8f czero = {};
    // D = A*B + 0  -> 16x16 tile of squared distances
    v8f d2 = __builtin_amdgcn_wmma_f32_16x16x4_f32(
        /*neg_a=*/false, afrag, /*neg_b=*/false, bfrag,
        /*c_mod=*/(short)0, czero, /*reuse_a=*/false, /*reuse_b=*/false);

    unsigned long long bestkey = 0ull;
#pragma unroll
    for (int r = 0; r < 8; ++r) {
      const float dd = fmaxf(d2[r], 0.0f);      // clamp fp rounding below zero
      rowmin[r] = fminf(rowmin[r], dd);
      // smaller d^2 (nonneg -> monotone uint bits) => larger key;
      // ties => lower pred index wins (matches jnp.argmax first-index rule)
      unsigned long long key =
          ((unsigned long long)(~__float_as_uint(dd)) << 32) |
          (unsigned long long)(keylo_base - (unsigned)r);
      // branchless validity: bitwise & (no short-circuit), cndmask to 0
      const bool valid = (dd <= R2_) & (pcls[r] == ctc);
      key = valid ? key : 0ull;
      bestkey = (key > bestkey) ? key : bestkey;   // v_max_u64
    }
    // lanes l and l^16 cover the same column (different row halves): combine
    unsigned long long other = __shfl_xor(bestkey, 16, 32);
    bestkey = (other > bestkey) ? other : bestkey;
    // lane>=16 holds a duplicate of the same column: only lanes 0-15 commit
    if ((lane < 16) & (bestkey != 0ull))
      atomicMax(&best[(size_t)b * T_ + (size_t)(t0 + idx)], bestkey);
  }

  // Reduce per-row d^2 min over the 16 columns, then ONE sigmoid per row.
#pragma unroll
  for (int r = 0; r < 8; ++r) {
    float v = rowmin[r];
    v = fminf(v, __shfl_xor(v, 8, 32));
    v = fminf(v, __shfl_xor(v, 4, 32));
    v = fminf(v, __shfl_xor(v, 2, 32));
    v = fminf(v, __shfl_xor(v, 1, 32));
    if (idx == 0) {
      const float d = __builtin_amdgcn_sqrtf(v + 1e-12f);             // v_sqrt_f32
      const float e = __builtin_amdgcn_exp2f((d - RADIUS_) * LOG2E_); // v_exp_f32
      const float s = __builtin_amdgcn_rcpf(1.0f + e);                // v_rcp_f32
      maxscore[(size_t)b * P_ + (size_t)(p0 + r + 8 * hf)] = s;
    }
  }
}

// ---------------------------------------------------------------------------
// Kernel 2: decode per-target winners -> correct[b][p*] = 1
// ---------------------------------------------------------------------------
__global__ void scatter_correct_kernel(const unsigned long long* __restrict__ best,
                                       int* __restrict__ correct) {
  int i = blockIdx.x * blockDim.x + threadIdx.x;
  if (i >= B_ * T_) return;
  unsigned long long k = best[i];
  if (k != 0ull) {
    int b = i / T_;
    unsigned p = 0xFFFFFFFFu - (unsigned)(k & 0xFFFFFFFFull);
    correct[(size_t)b * P_ + (size_t)p] = 1;   // idempotent store
  }
}

// ---------------------------------------------------------------------------
// Kernel 3: per-(b,p) loss terms + deterministic block tree-reduction
// ---------------------------------------------------------------------------
__global__ void loss_reduce_kernel(const float* __restrict__ pred,
                                   const int* __restrict__ correct,
                                   const float* __restrict__ maxscore,
                                   float* __restrict__ partials) {
  __shared__ float sobj[256];
  __shared__ float sreg[256];
  const int i = blockIdx.x * 256 + threadIdx.x;   // i = b*P + p, exactly B*P threads
  const float conf = pred[(size_t)i * 4 + 3];
  const int   c    = correct[i];
  const float t    = c ? 1.0f : 0.0f;
  // BCE with logits: max(x,0) - x*t + log1p(exp(-|x|))  (precise; only 65K elems)
  const float obj = fmaxf(conf, 0.0f) - conf * t + log1pf(expf(-fabsf(conf)));
  const float reg = 1.0f - (c ? maxscore[i] : 0.0f);
  sobj[threadIdx.x] = obj;
  sreg[threadIdx.x] = reg;
  __syncthreads();
  for (int s = 128; s > 0; s >>= 1) {
    if (threadIdx.x < (unsigned)s) {
      sobj[threadIdx.x] += sobj[threadIdx.x + s];
      sreg[threadIdx.x] += sreg[threadIdx.x + s];
    }
    __syncthreads();
  }
  if (threadIdx.x == 0) {
    partials[blockIdx.x * 2 + 0] = sobj[0];
    partials[blockIdx.x * 2 + 1] = sreg[0];
  }
}

// ---------------------------------------------------------------------------
// Kernel 4: final reduction of 256 partial pairs -> two scalars
// ---------------------------------------------------------------------------
__global__ void final_reduce_kernel(const float* __restrict__ partials,
                                    float* __restrict__ out) {
  __shared__ float sobj[256];
  __shared__ float sreg[256];
  const int t = threadIdx.x;
  sobj[t] = partials[t * 2 + 0];
  sreg[t] = partials[t * 2 + 1];
  __syncthreads();
  for (int s = 128; s > 0; s >>= 1) {
    if (t < s) { sobj[t] += sobj[t + s]; sreg[t] += sreg[t + s]; }
    __syncthreads();
  }
  if (t == 0) {
    const float inv = 1.0f / (float)(B_ * P_);
    out[0] = sobj[0] * inv;   // obj_loss
    out[1] = sreg[0] * inv;   // reg_loss
  }
}

// ---------------------------------------------------------------------------
// Host-side launch
// ---------------------------------------------------------------------------
extern "C" void kernel_launch(void* const* d_in, const int* in_sizes, int n_in,
                              void* d_out, int out_size, void* d_ws, size_t ws_size,
                              hipStream_t stream) {
  (void)in_sizes; (void)n_in; (void)out_size; (void)ws_size;
  const float* pred = (const float*)d_in[0];   // [16][4096][4]
  const float* gt   = (const float*)d_in[1];   // [16][1024][3]
  float* out        = (float*)d_out;           // 2 floats

  // Workspace layout
  char* ws = (char*)d_ws;
  unsigned long long* best = (unsigned long long*)(ws + 0);               // 128 KB
  float* maxscore          = (float*)(ws + 131072);                       // 256 KB
  int*   correct           = (int*)(ws + 131072 + 262144);                // 256 KB
  float* partials          = (float*)(ws + 131072 + 262144 + 262144);     //   2 KB

  // Zero best/maxscore/correct region: 655360 bytes = 163840 u32
  const int zwords = (131072 + 262144 + 262144) / 4;
  zero_ws_kernel<<<(zwords + 255) / 256, 256, 0, stream>>>((unsigned int*)d_ws, zwords);

  // 1 wave per (b, 16-pred tile): 16*256 = 4096 waves, 4 waves per 128-thread block
  const int nwaves = B_ * (P_ / 16);
  pair_kernel<<<nwaves / 4, 128, 0, stream>>>(pred, gt, best, maxscore);

  scatter_correct_kernel<<<(B_ * T_) / 256, 256, 0, stream>>>(best, correct);

  loss_reduce_kernel<<<(B_ * P_) / 256, 256, 0, stream>>>(pred, correct, maxscore, partials);

  final_reduce_kernel<<<1, 256, 0, stream>>>(partials, out);
}